// TwoAgentGNN_37589553775265
// MI455X (gfx1250) — compile-verified
//
#include <hip/hip_runtime.h>
#include <hip/hip_bf16.h>
#include <stdint.h>

#define DF 128  // feature dim for all layers (D_IN == D_HID == D_OUT == 128)

typedef __attribute__((ext_vector_type(2))) float v2f;
typedef __attribute__((ext_vector_type(8))) float v8f;

// ---------------------------------------------------------------------------
// Zero a float buffer (float4 vectorized).
// ---------------------------------------------------------------------------
__global__ void __launch_bounds__(256) zero_f32(float* __restrict__ p, int n4) {
  int i = blockIdx.x * blockDim.x + threadIdx.x;
  if (i < n4) {
    ((float4*)p)[i] = make_float4(0.f, 0.f, 0.f, 0.f);
  }
}

// ---------------------------------------------------------------------------
// Edge scatter-add: agg[dst[e]] += feat[src[e]]  (128 features per edge).
// One wave32 per edge: src/dst loads are wave-uniform (scalar), each lane
// moves a float4 => the wave gathers one contiguous 512B row from L2 and
// issues 4 global_atomic_add_f32 per lane. Feature tables are L2-resident
// (51.2MB << 192MB L2), so this runs at L2 gather/atomic speed.
// ---------------------------------------------------------------------------
__global__ void __launch_bounds__(256) scatter_add_edges(
    const float* __restrict__ feat, const int* __restrict__ src,
    const int* __restrict__ dst, float* __restrict__ agg, int n_edges) {
  int gid = blockIdx.x * blockDim.x + threadIdx.x;
  int e = gid >> 5;  // wave32: one edge per wave
  if (e >= n_edges) return;
  int lane = gid & 31;

  int s = src[e];  // uniform across the wave
  int d = dst[e];

  const float4 v = *(const float4*)(feat + (size_t)s * DF + lane * 4);
  float* ap = agg + (size_t)d * DF + lane * 4;
  atomicAdd(ap + 0, v.x);
  atomicAdd(ap + 1, v.y);
  atomicAdd(ap + 2, v.z);
  atomicAdd(ap + 3, v.w);
}

// ---------------------------------------------------------------------------
// Fused GraphConv epilogue GEMM:
//   out = act( agg @ Wrel + bias + xin @ Wroot )
// fp32 WMMA (V_WMMA_F32_16X16X4_F32). One 16x16 output tile per wave; 8
// waves/block cover all 128 output columns; blockIdx.x selects the 16-row
// band. Bias is pre-loaded into the accumulator (per-column, so identical in
// all 8 acc VGPRs).
//
// Fragment layouts (CDNA5 ISA 7.12.2, 32-bit, wave32):
//   A 16x4 : lane L<16 -> M=L, K={0,1}; lane L>=16 -> M=L-16, K={2,3}
//   B 4x16 : lane L<16 -> N=L, K={0,1}; lane L>=16 -> N=L-16, K={2,3}
//   C 16x16: VGPR v -> M = v + 8*(L>=16), N = L%16
// ---------------------------------------------------------------------------
__global__ void __launch_bounds__(256) gnn_gemm_wmma(
    const float* __restrict__ agg, const float* __restrict__ xin,
    const float* __restrict__ Wrel, const float* __restrict__ Wroot,
    const float* __restrict__ bias, float* __restrict__ out, int relu) {
  const int lane = threadIdx.x & 31;
  const int wave = threadIdx.x >> 5;
  const int row0 = blockIdx.x * 16;
  const int col0 = wave * 16;
  const int mr = lane & 15;   // A-row / B-col within tile
  const int half = lane >> 4; // selects K-pair within a K=4 step
  const int col = col0 + mr;

  const size_t arow = (size_t)(row0 + mr) * DF;

  const float bv = bias[col];
  v8f acc = {bv, bv, bv, bv, bv, bv, bv, bv};

  for (int k = 0; k < DF; k += 4) {
    const int kb = k + half * 2;
    // A fragments: two consecutive K values from row (row0+mr)
    v2f aA = *(const v2f*)(agg + arow + kb);
    v2f aX = *(const v2f*)(xin + arow + kb);
    // B fragments: W[kb][col], W[kb+1][col]  (W is [K=128][N=128] row-major)
    v2f bRel, bRoot;
    bRel.x  = Wrel [(size_t)kb * DF + col];
    bRel.y  = Wrel [(size_t)(kb + 1) * DF + col];
    bRoot.x = Wroot[(size_t)kb * DF + col];
    bRoot.y = Wroot[(size_t)(kb + 1) * DF + col];

    acc = __builtin_amdgcn_wmma_f32_16x16x4_f32(
        false, aA, false, bRel, (short)0, acc, false, false);
    acc = __builtin_amdgcn_wmma_f32_16x16x4_f32(
        false, aX, false, bRoot, (short)0, acc, false, false);
  }

#pragma unroll
  for (int v = 0; v < 8; ++v) {
    const int row = row0 + v + half * 8;
    float val = acc[v];
    if (relu) val = fmaxf(val, 0.f);
    out[(size_t)row * DF + col] = val;
  }
}

// ---------------------------------------------------------------------------
// Launch: zero->scatter->gemm(relu) -> zero->scatter->gemm
// ---------------------------------------------------------------------------
extern "C" void kernel_launch(void* const* d_in, const int* in_sizes, int n_in,
                              void* d_out, int out_size, void* d_ws,
                              size_t ws_size, hipStream_t stream) {
  const float* x      = (const float*)d_in[0];
  const int*   ei     = (const int*)d_in[1];
  const float* W1_rel = (const float*)d_in[2];
  const float* b1     = (const float*)d_in[3];
  const float* W1_root= (const float*)d_in[4];
  const float* W2_rel = (const float*)d_in[5];
  const float* b2     = (const float*)d_in[6];
  const float* W2_root= (const float*)d_in[7];

  const int nE = in_sizes[1] / 2;   // 1,600,000
  const int nN = in_sizes[0] / DF;  // 100,000
  const int* src = ei;              // edge_index[0]
  const int* dst = ei + nE;         // edge_index[1]

  float* agg = (float*)d_ws;                  // [nN, 128]
  float* h   = agg + (size_t)nN * DF;         // [nN, 128]
  float* out = (float*)d_out;

  const int n4    = nN * DF / 4;
  const int zgrid = (n4 + 255) / 256;
  const int sgrid = (int)(((size_t)nE * 32 + 255) / 256);
  const int ggrid = nN / 16;  // 100000 / 16 = 6250 exactly

  // Layer 1: h = relu(agg(x) @ W1_rel + b1 + x @ W1_root)
  zero_f32<<<zgrid, 256, 0, stream>>>(agg, n4);
  scatter_add_edges<<<sgrid, 256, 0, stream>>>(x, src, dst, agg, nE);
  gnn_gemm_wmma<<<ggrid, 256, 0, stream>>>(agg, x, W1_rel, W1_root, b1, h, 1);

  // Layer 2: out = agg(h) @ W2_rel + b2 + h @ W2_root
  zero_f32<<<zgrid, 256, 0, stream>>>(agg, n4);
  scatter_add_edges<<<sgrid, 256, 0, stream>>>(h, src, dst, agg, nE);
  gnn_gemm_wmma<<<ggrid, 256, 0, stream>>>(agg, h, W2_rel, W2_root, b2, out, 0);
}